// Tokenizer_29583734735474
// MI455X (gfx1250) — compile-verified
//
#include <hip/hip_runtime.h>
#include <hip/hip_bf16.h>

typedef float v2f __attribute__((ext_vector_type(2)));
typedef float v4f __attribute__((ext_vector_type(4)));
typedef float v8f __attribute__((ext_vector_type(8)));

#define BT   256                 // frames (b*t)
#define CH   512                 // z channels
#define HW   256                 // 16*16 spatial
#define EM   512                 // embed dim
#define VO   1024                // vocab
#define MROWS (BT*HW)            // 65536 spatial rows
#define NG   32                  // groups
#define CG   (CH/NG)             // 16 channels per group
#define GN_EPS 1e-6f

// -------------------------------------------------------------------------
// CDNA5 async global->LDS copy (ASYNCcnt-tracked, per-lane 16B)
// -------------------------------------------------------------------------
__device__ __forceinline__ void async_b128_to_lds(const float* gsrc, float* lds) {
  unsigned int      laddr = (unsigned int)(unsigned long long)(uintptr_t)lds;
  unsigned long long gaddr = (unsigned long long)(uintptr_t)gsrc;
  asm volatile("global_load_async_to_lds_b128 %0, %1, off"
               :: "v"(laddr), "v"(gaddr) : "memory");
}
__device__ __forceinline__ void wait_async0() {
  asm volatile("s_wait_asynccnt 0x0" ::: "memory");
}

// -------------------------------------------------------------------------
// Kernel 1: GroupNorm over [B,C,H,W], output transposed to rows [M, C]
// -------------------------------------------------------------------------
__global__ void gn_kernel(const float* __restrict__ x,
                          const float* __restrict__ gamma,
                          const float* __restrict__ beta,
                          float* __restrict__ xn_t) {
  __shared__ __align__(16) float s_tile[CG * HW];   // 16KB raw group data
  __shared__ float s_sum[256];
  __shared__ float s_sq[256];
  const int blk = blockIdx.x;
  const int b = blk >> 5, g = blk & 31;
  const int t = threadIdx.x;

  const float* xg = x + ((size_t)b * CH + (size_t)g * CG) * HW;
  float sum = 0.f, sq = 0.f;
#pragma unroll
  for (int i = 0; i < (CG * HW) / 256; ++i) {
    float v = xg[i * 256 + t];
    s_tile[i * 256 + t] = v;
    sum += v; sq += v * v;
  }
  s_sum[t] = sum; s_sq[t] = sq;
  __syncthreads();
  for (int s = 128; s > 0; s >>= 1) {
    if (t < s) { s_sum[t] += s_sum[t + s]; s_sq[t] += s_sq[t + s]; }
    __syncthreads();
  }
  const float inv_n = 1.0f / (float)(CG * HW);
  const float mu   = s_sum[0] * inv_n;
  const float var  = s_sq[0] * inv_n - mu * mu;
  const float rstd = rsqrtf(var + GN_EPS);

  // thread t owns pixel p = t; writes 16 consecutive channels (64B chunk)
  const int p = t;
  float* dst = xn_t + ((size_t)(b * HW + p)) * CH + g * CG;
#pragma unroll
  for (int cl = 0; cl < CG; ++cl) {
    const int c = g * CG + cl;
    dst[cl] = (s_tile[cl * HW + p] - mu) * rstd * gamma[c] + beta[c];
  }
}

// -------------------------------------------------------------------------
// Kernel 2: z = A[M,512] x w_pre[N=512,K=512]^T  via V_WMMA_F32_16X16X4_F32
// block = 256 thr (8 waves), M-tile = 32 rows (64KB LDS via async copy).
// Each wave: 2x2 blocking (2 m-subtiles x 2 n-tiles, 4 accumulators),
// so every k-step = 2 global b64 + 2 ds b64 -> 4 WMMAs.
// -------------------------------------------------------------------------
__global__ void gemm_pre_kernel(const float* __restrict__ A,
                                const float* __restrict__ W,
                                float* __restrict__ z_rows,
                                float* __restrict__ z_out) {
  __shared__ __align__(16) float sA[32 * 512];      // 64KB
  const int m0 = blockIdx.x * 32;
  const int t = threadIdx.x;
  {
    const float* src = A + (size_t)m0 * 512;
#pragma unroll
    for (int i = 0; i < 16; ++i) {
      const int fo = (i * 256 + t) * 4;
      async_b128_to_lds(src + fo, sA + fo);
    }
  }
  wait_async0();
  __syncthreads();

  const int lane = t & 31, wv = t >> 5;
  const int nloc  = lane & 15;
  const int khalf = (lane >> 4) << 1;           // 0 or 2
  const float* a0p = sA + (size_t)nloc * 512 + khalf;
  const float* a1p = a0p + 16 * 512;

#pragma unroll
  for (int j = 0; j < 2; ++j) {
    const int n0 = wv * 32 + j * 256;           // first n-tile base
    const int n1 = n0 + 16;                     // second n-tile base
    v8f acc00 = {}; v8f acc01 = {};
    v8f acc10 = {}; v8f acc11 = {};
    const float* w0 = W + (size_t)(n0 + nloc) * 512 + khalf;
    const float* w1 = W + (size_t)(n1 + nloc) * 512 + khalf;
#pragma unroll 4
    for (int k0 = 0; k0 < 512; k0 += 4) {
      v2f b0 = *(const v2f*)(w0 + k0);
      v2f b1 = *(const v2f*)(w1 + k0);
      v2f a0 = *(const v2f*)(a0p + k0);
      v2f a1 = *(const v2f*)(a1p + k0);
      acc00 = __builtin_amdgcn_wmma_f32_16x16x4_f32(false, a0, false, b0,
                                                    (short)0, acc00, false, false);
      acc01 = __builtin_amdgcn_wmma_f32_16x16x4_f32(false, a0, false, b1,
                                                    (short)0, acc01, false, false);
      acc10 = __builtin_amdgcn_wmma_f32_16x16x4_f32(false, a1, false, b0,
                                                    (short)0, acc10, false, false);
      acc11 = __builtin_amdgcn_wmma_f32_16x16x4_f32(false, a1, false, b1,
                                                    (short)0, acc11, false, false);
    }
    const int na = n0 + nloc, nb = n1 + nloc;
    const int mbase = (lane >> 4) << 3;          // 0 or 8
#pragma unroll
    for (int r = 0; r < 8; ++r) {
      const int m = m0 + mbase + r;
      const int m1 = m + 16;
      z_rows[(size_t)m  * 512 + na] = acc00[r];
      z_rows[(size_t)m  * 512 + nb] = acc01[r];
      z_rows[(size_t)m1 * 512 + na] = acc10[r];
      z_rows[(size_t)m1 * 512 + nb] = acc11[r];
      z_out[((size_t)(m  >> 8) * 512 + na) * 256 + (m  & 255)] = acc00[r];
      z_out[((size_t)(m  >> 8) * 512 + nb) * 256 + (m  & 255)] = acc01[r];
      z_out[((size_t)(m1 >> 8) * 512 + na) * 256 + (m1 & 255)] = acc10[r];
      z_out[((size_t)(m1 >> 8) * 512 + nb) * 256 + (m1 & 255)] = acc11[r];
    }
  }
}

// -------------------------------------------------------------------------
// Kernel 3: code norms ||e_v||^2 , one block per code
// -------------------------------------------------------------------------
__global__ void embnorm_kernel(const float* __restrict__ emb,
                               float* __restrict__ env) {
  __shared__ float s[256];
  const int v = blockIdx.x, t = threadIdx.x;
  const float a = emb[(size_t)v * 512 + t];
  const float b = emb[(size_t)v * 512 + 256 + t];
  s[t] = a * a + b * b;
  __syncthreads();
  for (int st = 128; st > 0; st >>= 1) {
    if (t < st) s[t] += s[t + st];
    __syncthreads();
  }
  if (t == 0) env[v] = s[0];
}

// -------------------------------------------------------------------------
// Kernel 4: fused distance-GEMM + argmin + codebook gather.
// score[m,v] = z.e_v - 0.5*||e_v||^2 ; argmax == argmin of L2 distance.
// block = 256 thr, M-tile = 16 rows (32KB LDS, async copy);
// each wave scans 128 codes as 4 tile-pairs sharing the A fragment.
// -------------------------------------------------------------------------
__global__ void vq_kernel(const float* __restrict__ z_rows,
                          const float* __restrict__ emb,
                          const float* __restrict__ env,
                          float* __restrict__ tok_out,
                          float* __restrict__ zq_rows,
                          float* __restrict__ zq_out) {
  __shared__ __align__(16) float sA[16 * 512];      // 32KB
  __shared__ float s_bv[8][16];
  __shared__ int   s_bi[8][16];
  __shared__ int   s_tok[16];
  const int m0 = blockIdx.x * 16;
  const int t = threadIdx.x;
  {
    const float* src = z_rows + (size_t)m0 * 512;
#pragma unroll
    for (int i = 0; i < 8; ++i) {
      const int fo = (i * 256 + t) * 4;
      async_b128_to_lds(src + fo, sA + fo);
    }
  }
  wait_async0();
  __syncthreads();

  const int lane = t & 31, wv = t >> 5;
  const int nloc  = lane & 15;
  const int khalf = (lane >> 4) << 1;
  const float* a0p = sA + (size_t)nloc * 512 + khalf;

  float bestv[8];
  int   besti[8];
#pragma unroll
  for (int r = 0; r < 8; ++r) { bestv[r] = -3.4e38f; besti[r] = 0x7fffffff; }

#pragma unroll
  for (int j = 0; j < 4; ++j) {
    const int vt0 = wv * 16 + j * 256;
    const int vt1 = vt0 + 128;
    v8f acc0 = {}; v8f acc1 = {};
    const int vc0 = vt0 + nloc, vc1 = vt1 + nloc;
    const float* e0 = emb + (size_t)vc0 * 512 + khalf;
    const float* e1 = emb + (size_t)vc1 * 512 + khalf;
#pragma unroll 4
    for (int k0 = 0; k0 < 512; k0 += 4) {
      v2f b0 = *(const v2f*)(e0 + k0);
      v2f b1 = *(const v2f*)(e1 + k0);
      v2f af = *(const v2f*)(a0p + k0);
      acc0 = __builtin_amdgcn_wmma_f32_16x16x4_f32(false, af, false, b0,
                                                   (short)0, acc0, false, false);
      acc1 = __builtin_amdgcn_wmma_f32_16x16x4_f32(false, af, false, b1,
                                                   (short)0, acc1, false, false);
    }
    const float en0 = env[vc0];
    const float en1 = env[vc1];
#pragma unroll
    for (int r = 0; r < 8; ++r) {
      const float s0 = acc0[r] - 0.5f * en0;
      if (s0 > bestv[r] || (s0 == bestv[r] && vc0 < besti[r])) {
        bestv[r] = s0; besti[r] = vc0;
      }
      const float s1 = acc1[r] - 0.5f * en1;
      if (s1 > bestv[r] || (s1 == bestv[r] && vc1 < besti[r])) {
        bestv[r] = s1; besti[r] = vc1;
      }
    }
  }

  // reduce over the 16 lanes of each half-wave (distinct m per half)
#pragma unroll
  for (int r = 0; r < 8; ++r) {
    float v = bestv[r]; int i = besti[r];
#pragma unroll
    for (int mask = 1; mask < 16; mask <<= 1) {
      float pv = __shfl_xor(v, mask, 32);
      int   pi = __shfl_xor(i, mask, 32);
      if (pv > v || (pv == v && pi < i)) { v = pv; i = pi; }
    }
    if ((lane & 15) == 0) {
      const int ml = ((lane >> 4) << 3) + r;
      s_bv[wv][ml] = v;
      s_bi[wv][ml] = i;
    }
  }
  __syncthreads();
  if (t < 16) {
    float v = s_bv[0][t]; int i = s_bi[0][t];
    for (int w = 1; w < 8; ++w) {
      const float pv = s_bv[w][t]; const int pi = s_bi[w][t];
      if (pv > v || (pv == v && pi < i)) { v = pv; i = pi; }
    }
    s_tok[t] = i;
    tok_out[m0 + t] = (float)i;
  }
  __syncthreads();

  // gather codebook rows -> zq_rows [M,E] and zq [B,E,H,W]
#pragma unroll
  for (int i = 0; i < 32; ++i) {
    const int flat = i * 256 + t;
    const int row = flat >> 9, e = flat & 511;
    const int tok = s_tok[row];
    const float val = emb[(size_t)tok * 512 + e];
    const int m = m0 + row;
    zq_rows[(size_t)m * 512 + e] = val;
    zq_out[((size_t)(m >> 8) * 512 + e) * 256 + (m & 255)] = val;
  }
}

// -------------------------------------------------------------------------
// Kernel 5: rec = zq_rows[M,512] x w_post[N=512,K=512]^T + bias  (WMMA f32)
// same 2x2 blocking as kernel 2
// -------------------------------------------------------------------------
__global__ void gemm_post_kernel(const float* __restrict__ A,
                                 const float* __restrict__ W,
                                 const float* __restrict__ bias,
                                 float* __restrict__ rec) {
  __shared__ __align__(16) float sA[32 * 512];      // 64KB
  const int m0 = blockIdx.x * 32;
  const int t = threadIdx.x;
  {
    const float* src = A + (size_t)m0 * 512;
#pragma unroll
    for (int i = 0; i < 16; ++i) {
      const int fo = (i * 256 + t) * 4;
      async_b128_to_lds(src + fo, sA + fo);
    }
  }
  wait_async0();
  __syncthreads();

  const int lane = t & 31, wv = t >> 5;
  const int nloc  = lane & 15;
  const int khalf = (lane >> 4) << 1;
  const float* a0p = sA + (size_t)nloc * 512 + khalf;
  const float* a1p = a0p + 16 * 512;

#pragma unroll
  for (int j = 0; j < 2; ++j) {
    const int n0 = wv * 32 + j * 256;
    const int n1 = n0 + 16;
    v8f acc00 = {}; v8f acc01 = {};
    v8f acc10 = {}; v8f acc11 = {};
    const float* w0 = W + (size_t)(n0 + nloc) * 512 + khalf;
    const float* w1 = W + (size_t)(n1 + nloc) * 512 + khalf;
#pragma unroll 4
    for (int k0 = 0; k0 < 512; k0 += 4) {
      v2f b0 = *(const v2f*)(w0 + k0);
      v2f b1 = *(const v2f*)(w1 + k0);
      v2f a0 = *(const v2f*)(a0p + k0);
      v2f a1 = *(const v2f*)(a1p + k0);
      acc00 = __builtin_amdgcn_wmma_f32_16x16x4_f32(false, a0, false, b0,
                                                    (short)0, acc00, false, false);
      acc01 = __builtin_amdgcn_wmma_f32_16x16x4_f32(false, a0, false, b1,
                                                    (short)0, acc01, false, false);
      acc10 = __builtin_amdgcn_wmma_f32_16x16x4_f32(false, a1, false, b0,
                                                    (short)0, acc10, false, false);
      acc11 = __builtin_amdgcn_wmma_f32_16x16x4_f32(false, a1, false, b1,
                                                    (short)0, acc11, false, false);
    }
    const int na = n0 + nloc, nb = n1 + nloc;
    const float bna = bias[na];
    const float bnb = bias[nb];
    const int mbase = (lane >> 4) << 3;
#pragma unroll
    for (int r = 0; r < 8; ++r) {
      const int m = m0 + mbase + r;
      const int m1 = m + 16;
      rec[((size_t)(m  >> 8) * 512 + na) * 256 + (m  & 255)] = acc00[r] + bna;
      rec[((size_t)(m  >> 8) * 512 + nb) * 256 + (m  & 255)] = acc01[r] + bnb;
      rec[((size_t)(m1 >> 8) * 512 + na) * 256 + (m1 & 255)] = acc10[r] + bna;
      rec[((size_t)(m1 >> 8) * 512 + nb) * 256 + (m1 & 255)] = acc11[r] + bnb;
    }
  }
}

// -------------------------------------------------------------------------
extern "C" void kernel_launch(void* const* d_in, const int* in_sizes, int n_in,
                              void* d_out, int out_size, void* d_ws, size_t ws_size,
                              hipStream_t stream) {
  const float* x      = (const float*)d_in[0];
  const float* gamma  = (const float*)d_in[1];
  const float* beta   = (const float*)d_in[2];
  const float* w_pre  = (const float*)d_in[3];
  const float* emb    = (const float*)d_in[4];
  const float* w_post = (const float*)d_in[5];
  const float* b_post = (const float*)d_in[6];

  float* out = (float*)d_out;
  float* z_out   = out;                               // [B,E,H,W]  33554432
  float* zq_out  = out + (size_t)33554432;            // [B,E,H,W]  33554432
  float* tok_out = out + (size_t)67108864;            // [B,HW]     65536
  float* rec_out = out + (size_t)67174400;            // [B,C,H,W]  33554432

  float* ws      = (float*)d_ws;
  float* xn_t    = ws;                                // [M,512]
  float* z_rows  = ws + (size_t)MROWS * 512;          // [M,512]
  float* env     = ws + (size_t)2 * MROWS * 512;      // [1024]
  float* zq_rows = xn_t;                              // reuse after GEMM1

  gn_kernel<<<BT * NG, 256, 0, stream>>>(x, gamma, beta, xn_t);
  gemm_pre_kernel<<<MROWS / 32, 256, 0, stream>>>(xn_t, w_pre, z_rows, z_out);
  embnorm_kernel<<<VO, 256, 0, stream>>>(emb, env);
  vq_kernel<<<MROWS / 16, 256, 0, stream>>>(z_rows, emb, env, tok_out, zq_rows, zq_out);
  gemm_post_kernel<<<MROWS / 32, 256, 0, stream>>>(zq_rows, w_post, b_post, rec_out);
}